// Model_73907797230059
// MI455X (gfx1250) — compile-verified
//
#include <hip/hip_runtime.h>
#include <hip/hip_bf16.h>
#include <math.h>

#define N_NODES 50000
#define N_EDGES 200000
#define F_INDIM 384
#define C_DIM   384
#define D_DIM   16
#define NEG_SLOPE 0.2f

typedef __attribute__((ext_vector_type(2))) float v2f;
typedef __attribute__((ext_vector_type(8))) float v8f;

__device__ __forceinline__ void atomicMaxFloat(float* addr, float val) {
    // monotonic bit-trick: int max for non-negative floats, uint min for negative
    if (val >= 0.0f) {
        atomicMax((int*)addr, __float_as_int(val));
    } else {
        atomicMin((unsigned int*)addr, __float_as_uint(val));
    }
}

// ---------------------------------------------------------------------------
// K1: tiny precompute (single block)
//   U_T[n][k] = (W_lin @ [att_src | att_dst | W_out])^T padded to 16 cols
//   vedge[d]  = W_edge[d,:] . att_edge
//   cout      = bias_conv . W_out + b_out
// ---------------------------------------------------------------------------
__global__ void prep_kernel(const float* __restrict__ W_lin,
                            const float* __restrict__ att_src,
                            const float* __restrict__ att_dst,
                            const float* __restrict__ W_edge,
                            const float* __restrict__ att_edge,
                            const float* __restrict__ bias_conv,
                            const float* __restrict__ W_out,
                            const float* __restrict__ b_out,
                            float* __restrict__ U_T,
                            float* __restrict__ vedge,
                            float* __restrict__ cout) {
    const int t = threadIdx.x;
    if (t < F_INDIM) {
        float s1 = 0.f, s2 = 0.f, s3 = 0.f;
        const float* row = W_lin + (size_t)t * C_DIM;
        for (int c = 0; c < C_DIM; ++c) {
            const float w = row[c];
            s1 += w * att_src[c];
            s2 += w * att_dst[c];
            s3 += w * W_out[c];
        }
        U_T[0 * F_INDIM + t] = s1;   // column 0: a_src
        U_T[1 * F_INDIM + t] = s2;   // column 1: a_dst
        U_T[2 * F_INDIM + t] = s3;   // column 2: p
        for (int n = 3; n < 16; ++n) U_T[n * F_INDIM + t] = 0.0f;
    } else if (t < F_INDIM + D_DIM) {
        const int d = t - F_INDIM;
        float s = 0.f;
        for (int c = 0; c < C_DIM; ++c) s += W_edge[(size_t)d * C_DIM + c] * att_edge[c];
        vedge[d] = s;
    } else if (t == F_INDIM + D_DIM) {
        float s = b_out[0];
        for (int c = 0; c < C_DIM; ++c) s += bias_conv[c] * W_out[c];
        cout[0] = s;
    }
}

// ---------------------------------------------------------------------------
// K2: init node accumulators
// ---------------------------------------------------------------------------
__global__ void init_kernel(float* __restrict__ deg, float* __restrict__ sdot,
                            float* __restrict__ amax, float* __restrict__ denom,
                            float* __restrict__ numacc, int n) {
    const int i = blockIdx.x * blockDim.x + threadIdx.x;
    if (i < n) {
        deg[i]    = 0.0f;
        sdot[i]   = 0.0f;
        amax[i]   = -INFINITY;
        denom[i]  = 0.0f;
        numacc[i] = 0.0f;
    }
}

// ---------------------------------------------------------------------------
// K3: skinny fp32 GEMM  [N,384] @ [384,16]  via V_WMMA_F32_16X16X4_F32.
// One wave per 16-row tile, K-loop of 96 WMMAs. fp32-exact (RNE FMA tree).
// A frag (16x4): lanes 0-15 rows M=lane, K = k0..k0+1; lanes 16-31 same rows,
//   K = k0+2..k0+3   -> float2 load per lane from x.
// B frag (4x16): VGPR j, lanes 0-15: K=k0+j; lanes 16-31: K=k0+2+j
//   -> float2 load from transposed U (U_T[n][k] contiguous in k).
// D (16x16 f32): VGPR j, lanes 0-15: M=j, lanes 16-31: M=8+j; col = lane&15.
// ---------------------------------------------------------------------------
__global__ __launch_bounds__(128) void matvec_wmma_kernel(
        const float* __restrict__ x,
        const float* __restrict__ U_T,
        float* __restrict__ Asrc,
        float* __restrict__ Adst,
        float* __restrict__ P) {
    const int lane  = threadIdx.x & 31;
    const int wave  = threadIdx.x >> 5;
    const int tile  = blockIdx.x * 4 + wave;
    const int ntile = N_NODES / 16;           // 3125, exact
    if (tile >= ntile) return;                // wave-uniform: EXEC stays all-1s

    const int khalf = (lane >> 4) << 1;       // 0 for lanes 0-15, 2 for 16-31
    const float* arow = x   + (size_t)(tile * 16 + (lane & 15)) * F_INDIM + khalf;
    const float* brow = U_T + (size_t)(lane & 15) * F_INDIM + khalf;

    v8f c = {0.f, 0.f, 0.f, 0.f, 0.f, 0.f, 0.f, 0.f};
    #pragma unroll 4
    for (int k0 = 0; k0 < F_INDIM; k0 += 4) {
        const v2f a = *(const v2f*)(arow + k0);
        const v2f b = *(const v2f*)(brow + k0);
        c = __builtin_amdgcn_wmma_f32_16x16x4_f32(
                /*neg_a=*/false, a, /*neg_b=*/false, b,
                /*c_mod=*/(short)0, c, /*reuse_a=*/false, /*reuse_b=*/false);
    }

    const int col   = lane & 15;
    const int mbase = tile * 16 + ((lane >> 4) << 3);
    if (col == 0) {
        #pragma unroll
        for (int j = 0; j < 8; ++j) Asrc[mbase + j] = c[j];
    } else if (col == 1) {
        #pragma unroll
        for (int j = 0; j < 8; ++j) Adst[mbase + j] = c[j];
    } else if (col == 2) {
        #pragma unroll
        for (int j = 0; j < 8; ++j) P[mbase + j] = c[j];
    }
}

// ---------------------------------------------------------------------------
// K4: per-edge a_edge dot (16) + degree / edge-attr-dot sums for self-loop mean
// ---------------------------------------------------------------------------
__global__ void edge_phase1(const int* __restrict__ dst,
                            const float* __restrict__ edge_attr,
                            const float* __restrict__ vedge,
                            float* __restrict__ aedge,
                            float* __restrict__ sdot,
                            float* __restrict__ deg) {
    const int e = blockIdx.x * blockDim.x + threadIdx.x;
    if (e >= N_EDGES) return;
    const float* ea = edge_attr + (size_t)e * D_DIM;
    float s = 0.f;
    #pragma unroll
    for (int i = 0; i < D_DIM; ++i) s += ea[i] * vedge[i];
    aedge[e] = s;
    const int d = dst[e];
    atomicAdd(&sdot[d], s);
    atomicAdd(&deg[d], 1.0f);
}

// ---------------------------------------------------------------------------
// K5: attention logits (leaky-relu) + segment max over destinations
// i < E: real edges ; i >= E: self loops (edge_attr = incoming mean)
// ---------------------------------------------------------------------------
__global__ void edge_phase2(const int* __restrict__ src,
                            const int* __restrict__ dst,
                            const float* __restrict__ Asrc,
                            const float* __restrict__ Adst,
                            float* __restrict__ aedge,
                            const float* __restrict__ sdot,
                            const float* __restrict__ deg,
                            float* __restrict__ aloop,
                            float* __restrict__ amax) {
    const int i = blockIdx.x * blockDim.x + threadIdx.x;
    if (i < N_EDGES) {
        const int s = src[i], d = dst[i];
        float al = Asrc[s] + Adst[d] + aedge[i];
        al = (al > 0.f) ? al : NEG_SLOPE * al;
        aedge[i] = al;
        atomicMaxFloat(&amax[d], al);
    } else if (i < N_EDGES + N_NODES) {
        const int n = i - N_EDGES;
        const float m = sdot[n] / fmaxf(deg[n], 1.0f);
        float al = Asrc[n] + Adst[n] + m;
        al = (al > 0.f) ? al : NEG_SLOPE * al;
        aloop[n] = al;
        atomicMaxFloat(&amax[n], al);
    }
}

// ---------------------------------------------------------------------------
// K6: softmax exp + denominator + weighted numerator (scalar message p[src])
// ---------------------------------------------------------------------------
__global__ void edge_phase3(const int* __restrict__ src,
                            const int* __restrict__ dst,
                            const float* __restrict__ aedge,
                            const float* __restrict__ aloop,
                            const float* __restrict__ amax,
                            const float* __restrict__ P,
                            float* __restrict__ denom,
                            float* __restrict__ numacc) {
    const int i = blockIdx.x * blockDim.x + threadIdx.x;
    if (i < N_EDGES) {
        const int s = src[i], d = dst[i];
        const float ex = expf(aedge[i] - amax[d]);
        atomicAdd(&denom[d], ex);
        atomicAdd(&numacc[d], ex * P[s]);
    } else if (i < N_EDGES + N_NODES) {
        const int n = i - N_EDGES;
        const float ex = expf(aloop[n] - amax[n]);
        atomicAdd(&denom[n], ex);
        atomicAdd(&numacc[n], ex * P[n]);
    }
}

// ---------------------------------------------------------------------------
// K7: out = relu(num/denom + (bias_conv . W_out + b_out))
// ---------------------------------------------------------------------------
__global__ void final_kernel(const float* __restrict__ numacc,
                             const float* __restrict__ denom,
                             const float* __restrict__ cout,
                             float* __restrict__ out, int n) {
    const int i = blockIdx.x * blockDim.x + threadIdx.x;
    if (i < n) {
        const float v = numacc[i] / denom[i] + cout[0];
        out[i] = fmaxf(v, 0.0f);
    }
}

extern "C" void kernel_launch(void* const* d_in, const int* in_sizes, int n_in,
                              void* d_out, int out_size, void* d_ws, size_t ws_size,
                              hipStream_t stream) {
    const float* x          = (const float*)d_in[0];
    const int*   edge_index = (const int*)d_in[1];     // [2, E]
    const float* edge_attr  = (const float*)d_in[2];
    const float* W_lin      = (const float*)d_in[3];
    const float* att_src    = (const float*)d_in[4];
    const float* att_dst    = (const float*)d_in[5];
    const float* W_edge     = (const float*)d_in[6];
    const float* att_edge   = (const float*)d_in[7];
    const float* bias_conv  = (const float*)d_in[8];
    const float* W_out      = (const float*)d_in[9];
    const float* b_out      = (const float*)d_in[10];
    float* out = (float*)d_out;

    const int* src = edge_index;
    const int* dst = edge_index + N_EDGES;

    // workspace carve (floats): ~2.6 MB total
    float* ws     = (float*)d_ws;
    float* U_T    = ws;                       // 16*384 = 6144
    float* vedge  = U_T + 16 * F_INDIM;       // 16
    float* cout   = vedge + 16;               // 1
    float* Asrc   = ws + 8192;                // N
    float* Adst   = Asrc + N_NODES;           // N
    float* P      = Adst + N_NODES;           // N
    float* deg    = P + N_NODES;              // N
    float* sdot   = deg + N_NODES;            // N
    float* amax   = sdot + N_NODES;           // N
    float* denom  = amax + N_NODES;           // N
    float* numacc = denom + N_NODES;          // N
    float* aedge  = numacc + N_NODES;         // E
    float* aloop  = aedge + N_EDGES;          // N

    prep_kernel<<<1, 416, 0, stream>>>(W_lin, att_src, att_dst, W_edge, att_edge,
                                       bias_conv, W_out, b_out, U_T, vedge, cout);

    init_kernel<<<(N_NODES + 255) / 256, 256, 0, stream>>>(deg, sdot, amax, denom,
                                                           numacc, N_NODES);

    const int ntiles = N_NODES / 16;          // 3125
    matvec_wmma_kernel<<<(ntiles + 3) / 4, 128, 0, stream>>>(x, U_T, Asrc, Adst, P);

    edge_phase1<<<(N_EDGES + 255) / 256, 256, 0, stream>>>(dst, edge_attr, vedge,
                                                           aedge, sdot, deg);

    const int m = N_EDGES + N_NODES;
    edge_phase2<<<(m + 255) / 256, 256, 0, stream>>>(src, dst, Asrc, Adst, aedge,
                                                     sdot, deg, aloop, amax);

    edge_phase3<<<(m + 255) / 256, 256, 0, stream>>>(src, dst, aedge, aloop, amax,
                                                     P, denom, numacc);

    final_kernel<<<(N_NODES + 255) / 256, 256, 0, stream>>>(numacc, denom, cout,
                                                            out, N_NODES);
}